// GAT_weight_47442208751841
// MI455X (gfx1250) — compile-verified
//
#include <hip/hip_runtime.h>
#include <hip/hip_bf16.h>
#include <math.h>

// ---------------------------------------------------------------------------
// GAT (2-layer, PyG-style) for MI455X / gfx1250, wave32.
// GEMMs use V_WMMA_F32_16X16X4_F32 (fp32 WMMA) to match fp32 reference math.
// Edge softmax/aggregation via float atomics (L2-resident working set).
// ---------------------------------------------------------------------------

typedef float v2f __attribute__((ext_vector_type(2)));
typedef float v8f __attribute__((ext_vector_type(8)));

#define NNODES   20000
#define NEDGES0  320000
#define IN_CH    256
#define HID      32
#define HEADS    8
#define NCLASSES 16
#define NEG_SLOPE 0.2f
#define EPS_DEN  1e-16f

// --------------------------- fp32 WMMA GEMM --------------------------------
// C[M,N] = A[M,K] * B[K,N], all row-major. Requires M%16==0, N%16==0, K%4==0.
// One wave32 computes one 16x16 tile of C; K-loop in steps of 4 using
// v_wmma_f32_16x16x4_f32.
__global__ __launch_bounds__(256) void wmma_gemm_f32(
    const float* __restrict__ A, const float* __restrict__ B,
    float* __restrict__ C, int M, int N, int K) {
  const int wavesPerBlock = blockDim.x >> 5;
  const int waveId = blockIdx.x * wavesPerBlock + (threadIdx.x >> 5);
  const int lane   = threadIdx.x & 31;
  const int tilesN = N >> 4;
  const int totalTiles = (M >> 4) * tilesN;
  if (waveId >= totalTiles) return;              // wave-uniform exit (EXEC all-1 for WMMA)
  const int mt = waveId / tilesN;
  const int nt = waveId % tilesN;

  const int r    = lane & 15;   // M-row within tile (A), N-col within tile (B/C)
  const int half = lane >> 4;   // 0: K+0/K+1 ; 1: K+2/K+3 (A,B); C rows +8

  const float* __restrict__ Arow = A + (size_t)(mt * 16 + r) * K;
  const float* __restrict__ Bcol = B + (nt * 16 + r);

  v8f c = {};
  for (int k0 = 0; k0 < K; k0 += 4) {
    v2f a, b;
    const int ka = k0 + 2 * half;
    a.x = Arow[ka];
    a.y = Arow[ka + 1];
    b.x = Bcol[(size_t)ka * N];
    b.y = Bcol[(size_t)(ka + 1) * N];
    c = __builtin_amdgcn_wmma_f32_16x16x4_f32(
        /*neg_a=*/false, a, /*neg_b=*/false, b,
        /*c_mod=*/(short)0, c, /*reuse_a=*/false, /*reuse_b=*/false);
  }

  const int col     = nt * 16 + r;
  const int rowBase = mt * 16 + half * 8;        // lanes 16-31 hold M=8..15
#pragma unroll
  for (int i = 0; i < 8; ++i)
    C[(size_t)(rowBase + i) * N + col] = c[i];
}

// --------------------------- helpers ---------------------------------------
__global__ void fill_f32(float* __restrict__ p, float v, int n) {
  int t = blockIdx.x * blockDim.x + threadIdx.x;
  if (t < n) p[t] = v;
}

__device__ inline void atomicMaxF32(float* addr, float val) {
  unsigned int* ai = (unsigned int*)addr;
  unsigned int old = *ai;
  while (__uint_as_float(old) < val) {
    unsigned int assumed = old;
    old = atomicCAS(ai, assumed, __float_as_uint(val));
    if (old == assumed) break;
  }
}

// a_s[n,h] = dot(h[n,h,:], att_src[h,:]) ; same for a_d.
__global__ void attn_scores(const float* __restrict__ h,
                            const float* __restrict__ att_src,
                            const float* __restrict__ att_dst,
                            float* __restrict__ a_s, float* __restrict__ a_d,
                            int Nn, int H, int F) {
  int t = blockIdx.x * blockDim.x + threadIdx.x;
  if (t >= Nn * H) return;
  const int n = t / H, hd = t % H;
  const float* hp = h + (size_t)n * H * F + (size_t)hd * F;
  const float* as = att_src + hd * F;
  const float* ad = att_dst + hd * F;
  float s = 0.f, d = 0.f;
  for (int f = 0; f < F; ++f) { float v = hp[f]; s += v * as[f]; d += v * ad[f]; }
  a_s[t] = s; a_d[t] = d;
}

// Pass 1: logit = leaky_relu(a_s[src]+a_d[dst]) * ew ; segment-max into m[dst].
__global__ void edge_logits_max(const int* __restrict__ src,
                                const int* __restrict__ dst,
                                const float* __restrict__ ew,
                                const float* __restrict__ a_s,
                                const float* __restrict__ a_d,
                                float* __restrict__ logit,
                                float* __restrict__ m,
                                int E0, int Nn, int H) {
  long long t = (long long)blockIdx.x * blockDim.x + threadIdx.x;
  long long total = (long long)(E0 + Nn) * H;
  if (t >= total) return;
  const int e = (int)(t / H), hd = (int)(t % H);
  int s, d; float w;
  if (e < E0) { s = src[e]; d = dst[e]; w = ew[e]; }
  else        { s = d = e - E0; w = 1.0f; }            // self-loop, weight 1
  float x = a_s[(size_t)s * H + hd] + a_d[(size_t)d * H + hd];
  x = (x > 0.f) ? x : NEG_SLOPE * x;
  x *= w;
  logit[t] = x;
  atomicMaxF32(&m[(size_t)d * H + hd], x);
}

// Pass 2: ex = exp(logit - m[dst]) ; segment-sum into denom[dst]. (in-place)
__global__ void edge_exp_sum(const int* __restrict__ dst,
                             float* __restrict__ logit,
                             const float* __restrict__ m,
                             float* __restrict__ denom,
                             int E0, int Nn, int H) {
  long long t = (long long)blockIdx.x * blockDim.x + threadIdx.x;
  long long total = (long long)(E0 + Nn) * H;
  if (t >= total) return;
  const int e = (int)(t / H), hd = (int)(t % H);
  const int d = (e < E0) ? dst[e] : (e - E0);
  float ex = __expf(logit[t] - m[(size_t)d * H + hd]);
  logit[t] = ex;
  atomicAdd(&denom[(size_t)d * H + hd], ex);
}

// Pass 3: out[dst,h,f] += (ex/denom[dst,h]) * hfeat[src,h,f].
__global__ void edge_aggregate(const int* __restrict__ src,
                               const int* __restrict__ dst,
                               const float* __restrict__ ex,
                               const float* __restrict__ denom,
                               const float* __restrict__ hfeat,
                               float* __restrict__ out,
                               int E0, int Nn, int H, int F) {
  long long t = (long long)blockIdx.x * blockDim.x + threadIdx.x;
  long long total = (long long)(E0 + Nn) * H * F;
  if (t >= total) return;
  const int f = (int)(t % F);
  long long eh = t / F;
  const int hd = (int)(eh % H);
  const int e  = (int)(eh / H);
  int s, d;
  if (e < E0) { s = src[e]; d = dst[e]; } else { s = d = e - E0; }
  const float alpha = ex[eh] / (denom[(size_t)d * H + hd] + EPS_DEN);
  const float msg = alpha * hfeat[((size_t)s * H + hd) * F + f];
  atomicAdd(&out[((size_t)d * H + hd) * F + f], msg);
}

// out[n,c] = (out[n,c] + bias[c]) ; optional ELU.
__global__ void bias_act(float* __restrict__ io, const float* __restrict__ bias,
                         int Nn, int C, int doElu) {
  int t = blockIdx.x * blockDim.x + threadIdx.x;
  if (t >= Nn * C) return;
  float v = io[t] + bias[t % C];
  if (doElu) v = (v > 0.f) ? v : (__expf(v) - 1.0f);
  io[t] = v;
}

// ---------------------------------------------------------------------------
extern "C" void kernel_launch(void* const* d_in, const int* in_sizes, int n_in,
                              void* d_out, int out_size, void* d_ws, size_t ws_size,
                              hipStream_t stream) {
  const float* x        = (const float*)d_in[0];
  const int*   eidx     = (const int*)d_in[1];
  const float* ew       = (const float*)d_in[2];
  const float* W1       = (const float*)d_in[3];
  const float* att_src1 = (const float*)d_in[4];
  const float* att_dst1 = (const float*)d_in[5];
  const float* b1       = (const float*)d_in[6];
  const float* W2       = (const float*)d_in[7];
  const float* att_src2 = (const float*)d_in[8];
  const float* att_dst2 = (const float*)d_in[9];
  const float* b2       = (const float*)d_in[10];
  float* out = (float*)d_out;

  const int* srcI = eidx;            // edge_index[0]
  const int* dstI = eidx + NEDGES0;  // edge_index[1]

  const int N  = NNODES;
  const int E0 = NEDGES0;
  const int ET = E0 + N;             // with self-loops
  const int C1 = HEADS * HID;        // 256

  // ---- workspace carve-up (floats) ----
  float* ws = (float*)d_ws;
  size_t off = 0;
  float* h1     = ws + off; off += (size_t)N * C1;        // [N,256]
  float* as1    = ws + off; off += (size_t)N * HEADS;     // [N,8]
  float* ad1    = ws + off; off += (size_t)N * HEADS;
  float* m1     = ws + off; off += (size_t)N * HEADS;
  float* den1   = ws + off; off += (size_t)N * HEADS;
  float* elog1  = ws + off; off += (size_t)ET * HEADS;    // [ET,8]
  float* out1   = ws + off; off += (size_t)N * C1;        // [N,256]
  float* h2     = ws + off; off += (size_t)N * NCLASSES;  // [N,16]
  float* as2    = ws + off; off += (size_t)N;
  float* ad2    = ws + off; off += (size_t)N;
  float* m2     = ws + off; off += (size_t)N;
  float* den2   = ws + off; off += (size_t)N;
  float* elog2  = ws + off; off += (size_t)ET;            // [ET,1]

  const int B = 256;
  auto blocks  = [](long long n, int b) { return (unsigned)((n + b - 1) / b); };

  // =================== Layer 1 ===================
  // h1 = x @ W1  (20000x256x256) — WMMA f32
  {
    int tiles = (N / 16) * (C1 / 16);
    wmma_gemm_f32<<<blocks(tiles, 8), 256, 0, stream>>>(x, W1, h1, N, C1, IN_CH);
  }
  attn_scores<<<blocks((long long)N * HEADS, B), B, 0, stream>>>(
      h1, att_src1, att_dst1, as1, ad1, N, HEADS, HID);

  fill_f32<<<blocks((long long)N * HEADS, B), B, 0, stream>>>(m1, -INFINITY, N * HEADS);
  fill_f32<<<blocks((long long)N * HEADS, B), B, 0, stream>>>(den1, 0.f, N * HEADS);
  fill_f32<<<blocks((long long)N * C1, B), B, 0, stream>>>(out1, 0.f, N * C1);

  edge_logits_max<<<blocks((long long)ET * HEADS, B), B, 0, stream>>>(
      srcI, dstI, ew, as1, ad1, elog1, m1, E0, N, HEADS);
  edge_exp_sum<<<blocks((long long)ET * HEADS, B), B, 0, stream>>>(
      dstI, elog1, m1, den1, E0, N, HEADS);
  edge_aggregate<<<blocks((long long)ET * HEADS * HID, B), B, 0, stream>>>(
      srcI, dstI, elog1, den1, h1, out1, E0, N, HEADS, HID);

  bias_act<<<blocks((long long)N * C1, B), B, 0, stream>>>(out1, b1, N, C1, /*elu=*/1);

  // =================== Layer 2 ===================
  // h2 = out1 @ W2  (20000x256x16) — WMMA f32
  {
    int tiles = (N / 16) * (NCLASSES / 16);
    wmma_gemm_f32<<<blocks(tiles, 8), 256, 0, stream>>>(out1, W2, h2, N, NCLASSES, C1);
  }
  attn_scores<<<blocks((long long)N, B), B, 0, stream>>>(
      h2, att_src2, att_dst2, as2, ad2, N, 1, NCLASSES);

  fill_f32<<<blocks((long long)N, B), B, 0, stream>>>(m2, -INFINITY, N);
  fill_f32<<<blocks((long long)N, B), B, 0, stream>>>(den2, 0.f, N);
  fill_f32<<<blocks((long long)N * NCLASSES, B), B, 0, stream>>>(out, 0.f, N * NCLASSES);

  edge_logits_max<<<blocks((long long)ET, B), B, 0, stream>>>(
      srcI, dstI, ew, as2, ad2, elog2, m2, E0, N, 1);
  edge_exp_sum<<<blocks((long long)ET, B), B, 0, stream>>>(
      dstI, elog2, m2, den2, E0, N, 1);
  edge_aggregate<<<blocks((long long)ET * NCLASSES, B), B, 0, stream>>>(
      srcI, dstI, elog2, den2, h2, out, E0, N, 1, NCLASSES);

  // concat=False with H=1 -> mean over 1 head == identity; just add bias.
  bias_act<<<blocks((long long)N * NCLASSES, B), B, 0, stream>>>(out, b2, N, NCLASSES, /*elu=*/0);
}